// GCNLayer_35802847380161
// MI455X (gfx1250) — compile-verified
//
#include <hip/hip_runtime.h>

typedef float v2f __attribute__((ext_vector_type(2)));
typedef float v8f __attribute__((ext_vector_type(8)));

#define D 128   // D_IN == D_OUT == 128

// -----------------------------------------------------------------------------
// Kernel 1: support = x @ W using V_WMMA_F32_16X16X4_F32 (exact fp32 path).
// Block = 256 threads = 8 waves. Block b computes rows [16b, 16b+16); wave w
// computes the 16x16 output tile at columns [16w, 16w+16). K-loop steps by 4.
//
// Fragment layouts (CDNA5 ISA 7.12.2, 32-bit data):
//   A 16x4 : lanes 0-15 -> {K=0,K=1} for M=lane ; lanes 16-31 -> {K=2,K=3}
//   B 4x16 : lanes 0-15 -> {K=0,K=1} for N=lane ; lanes 16-31 -> {K=2,K=3}
//   C 16x16: VGPR i -> M = i (lanes 0-15) / M = i+8 (lanes 16-31), N = lane%16
// -----------------------------------------------------------------------------
__global__ __launch_bounds__(256) void gcn_gemm_wmma(
    const float* __restrict__ x, const float* __restrict__ w,
    float* __restrict__ support, int n_nodes) {
  const int lane = threadIdx.x & 31;
  const int wave = threadIdx.x >> 5;
  const int m    = lane & 15;
  const int half = lane >> 4;          // 0 or 1 (selects K pair / M half)
  const int row0 = blockIdx.x * 16;
  const int col0 = wave * 16;

  int arow = row0 + m;
  if (arow >= n_nodes) arow = n_nodes - 1;   // clamp: keep EXEC all-1s for WMMA
  const float* xrow = x + (size_t)arow * D;
  const int n = col0 + m;                    // this lane's B/C column

  v8f acc = {};
#pragma unroll 4
  for (int k = 0; k < D; k += 4) {
    const int ka = k + half * 2;
    v2f a = *(const v2f*)(xrow + ka);        // 8B-aligned global_load_b64
    v2f b;
    b.x = w[(size_t)(ka + 0) * D + n];
    b.y = w[(size_t)(ka + 1) * D + n];
    acc = __builtin_amdgcn_wmma_f32_16x16x4_f32(
        /*neg_a=*/false, a, /*neg_b=*/false, b,
        /*c_mod=*/(short)0, acc, /*reuse_a=*/false, /*reuse_b=*/false);
  }

  // Epilogue: uniform full-tile test -> straight-line stores in the hot path
  // (N_NODES % 16 == 0, so the guarded path is never taken in practice).
  float* srow = support + (size_t)(row0 + half * 8) * D + n;
  if (row0 + 16 <= n_nodes) {
#pragma unroll
    for (int i = 0; i < 8; ++i) srow[(size_t)i * D] = acc[i];
  } else {
#pragma unroll
    for (int i = 0; i < 8; ++i) {
      if (row0 + half * 8 + i < n_nodes) srow[(size_t)i * D] = acc[i];
    }
  }
}

// -----------------------------------------------------------------------------
// Kernel 2: out[i][j] = bias[j]  (must re-init every call; harness poisons once)
// -----------------------------------------------------------------------------
__global__ __launch_bounds__(256) void gcn_init_out(
    float* __restrict__ out, const float* __restrict__ bias, int total) {
  int i = blockIdx.x * blockDim.x + threadIdx.x;
  if (i < total) out[i] = bias[i & (D - 1)];
}

// -----------------------------------------------------------------------------
// Kernel 3: out[adj_row[e]] += adj_val[e] * support[adj_col[e]]
// One wave per edge; lanes stream 128 contiguous floats (4 coalesced passes).
// adj_* loads are wave-uniform -> scalarized. support/out live in L2 (41 MB
// combined << 192 MB L2), so gathers and f32 atomics stay on-chip.
// -----------------------------------------------------------------------------
__global__ __launch_bounds__(256) void gcn_spmm_scatter(
    const int* __restrict__ adj_row, const int* __restrict__ adj_col,
    const float* __restrict__ adj_val, const float* __restrict__ support,
    float* __restrict__ out, int n_edges) {
  const int lane = threadIdx.x & 31;
  const int e = blockIdx.x * (blockDim.x >> 5) + (threadIdx.x >> 5);
  if (e >= n_edges) return;

  const int src  = adj_col[e];
  const int dst  = adj_row[e];
  const float v  = adj_val[e];
  const float* srow = support + (size_t)src * D;
  float* drow       = out + (size_t)dst * D;

#pragma unroll
  for (int j = 0; j < D; j += 32) {
    atomicAdd(drow + j + lane, v * srow[j + lane]);
  }
}

// -----------------------------------------------------------------------------
// Launch: inputs are (x, adj_row, adj_col, adj_val, weight, bias) per reference.
// d_ws holds support [n_nodes x 128] fp32 (20.5 MB).
// -----------------------------------------------------------------------------
extern "C" void kernel_launch(void* const* d_in, const int* in_sizes, int n_in,
                              void* d_out, int out_size, void* d_ws, size_t ws_size,
                              hipStream_t stream) {
  const float* x       = (const float*)d_in[0];
  const int*   adj_row = (const int*)d_in[1];
  const int*   adj_col = (const int*)d_in[2];
  const float* adj_val = (const float*)d_in[3];
  const float* weight  = (const float*)d_in[4];
  const float* bias    = (const float*)d_in[5];
  float* out     = (float*)d_out;
  float* support = (float*)d_ws;

  const int n_nodes = in_sizes[0] / D;
  const int n_edges = in_sizes[1];
  const int total   = n_nodes * D;

  gcn_gemm_wmma<<<(n_nodes + 15) / 16, 256, 0, stream>>>(x, weight, support, n_nodes);
  gcn_init_out<<<(total + 255) / 256, 256, 0, stream>>>(out, bias, total);
  gcn_spmm_scatter<<<(n_edges + 7) / 8, 256, 0, stream>>>(adj_row, adj_col, adj_val,
                                                          support, out, n_edges);
}